// Pose_Loss_40802189312414
// MI455X (gfx1250) — compile-verified
//
#include <hip/hip_runtime.h>

// ---------------------------------------------------------------------------
// Pose loss: masked row-sum reduction over four [B=65536, D=154] f32 matrices.
// Bandwidth-bound (161.5 MB @ 23.3 TB/s ~= 7 us). Row sums are computed on the
// matrix pipe with V_WMMA_F32_16X16X4_F32 against an all-ones B matrix so the
// VALU stays free for address generation / load issue.
// ---------------------------------------------------------------------------

typedef float v2f __attribute__((ext_vector_type(2)));
typedef float v8f __attribute__((ext_vector_type(8)));

#define D_DIM            154
#define D_WMMA           152   // 38 * 4, columns handled by WMMA; 2-col tail scalar
#define ROWS_PER_WAVE    16
#define WAVES_PER_BLOCK  4
#define BLOCK_THREADS    (WAVES_PER_BLOCK * 32)
#define ROWS_PER_BLOCK   (ROWS_PER_WAVE * WAVES_PER_BLOCK)

__device__ __forceinline__ void pose_weights(int lrv, int fpv, float& wl, float& wr) {
    const bool valid = (fpv != -1);
    wl = (valid && ((lrv == 0) || (lrv == 2))) ? 1.0f : 0.0f;
    wr = (valid && ((lrv == 1) || (lrv == 2))) ? 1.0f : 0.0f;
}

__global__ __launch_bounds__(BLOCK_THREADS)
void pose_loss_stage1(const float* __restrict__ rl, const float* __restrict__ rr,
                      const float* __restrict__ kl, const float* __restrict__ kr,
                      const int* __restrict__ lr, const int* __restrict__ fp,
                      float* __restrict__ ws, int nrows) {
    const int tid  = threadIdx.x;
    const int wave = tid >> 5;
    const int lane = tid & 31;
    const int m    = lane & 15;   // A-fragment row index (M = lane % 16)
    const int half = lane >> 4;   // K pair selector (0 -> K=0,1 ; 1 -> K=2,3)

    const long long R0 = ((long long)blockIdx.x * WAVES_PER_BLOCK + wave) * ROWS_PER_WAVE;
    const bool active = (R0 + ROWS_PER_WAVE) <= (long long)nrows;  // wave-uniform

    float contrib_r = 0.0f, contrib_k = 0.0f, contrib_c = 0.0f;

    if (active) {
        // A-fragment layout for V_WMMA_F32_16X16X4_F32 (wave32):
        //   VGPR0: lanes0-15 -> (M=m, K=0), lanes16-31 -> (M=m, K=2)
        //   VGPR1: lanes0-15 -> (M=m, K=1), lanes16-31 -> (M=m, K=3)
        // => per lane one float2 at row*D + k0 + 2*half (8B aligned: all terms even).
        const size_t base = (size_t)(R0 + m) * D_DIM + (size_t)(half << 1);

        const v2f ones = {1.0f, 1.0f};        // B = 4x16 all-ones => D[m][n] = rowsum(A)
        v8f c_rl = {0,0,0,0,0,0,0,0};
        v8f c_rr = {0,0,0,0,0,0,0,0};
        v8f c_kl = {0,0,0,0,0,0,0,0};
        v8f c_kr = {0,0,0,0,0,0,0,0};

        for (int k0 = 0; k0 < D_WMMA; k0 += 4) {
            if ((k0 & 31) == 0) {  // gfx1250 global_prefetch_b8, ~64 floats ahead
                __builtin_prefetch(rl + base + k0 + 64, 0, 1);
                __builtin_prefetch(rr + base + k0 + 64, 0, 1);
                __builtin_prefetch(kl + base + k0 + 64, 0, 1);
                __builtin_prefetch(kr + base + k0 + 64, 0, 1);
            }
            const v2f a_rl = *(const v2f*)(rl + base + k0);
            const v2f a_rr = *(const v2f*)(rr + base + k0);
            const v2f a_kl = *(const v2f*)(kl + base + k0);
            const v2f a_kr = *(const v2f*)(kr + base + k0);
            c_rl = __builtin_amdgcn_wmma_f32_16x16x4_f32(false, a_rl, false, ones,
                                                         (short)0, c_rl, false, false);
            c_rr = __builtin_amdgcn_wmma_f32_16x16x4_f32(false, a_rr, false, ones,
                                                         (short)0, c_rr, false, false);
            c_kl = __builtin_amdgcn_wmma_f32_16x16x4_f32(false, a_kl, false, ones,
                                                         (short)0, c_kl, false, false);
            c_kr = __builtin_amdgcn_wmma_f32_16x16x4_f32(false, a_kr, false, ones,
                                                         (short)0, c_kr, false, false);
        }

        // C/D layout: VGPR j = row (j) on lanes0-15, row (j+8) on lanes16-31;
        // every column n holds the same row sum (B was all ones). Lane 0 of each
        // half applies the mask weights for its 8 rows.
        if (m == 0) {
            #pragma unroll
            for (int j = 0; j < 8; ++j) {
                const long long rj = R0 + (long long)(half * 8 + j);
                float wl, wr;
                pose_weights(lr[rj], fp[rj], wl, wr);
                contrib_r += wl * c_rl[j] + wr * c_rr[j];
                contrib_k += wl * c_kl[j] + wr * c_kr[j];
                contrib_c += wl + wr;
            }
        }
        // Scalar tail (columns 152, 153): lane m (lanes 0-15) owns row R0+m.
        if (half == 0) {
            const long long rj  = R0 + m;
            const size_t    off = (size_t)rj * D_DIM + D_WMMA;
            float wl, wr;
            pose_weights(lr[rj], fp[rj], wl, wr);
            contrib_r += wl * (rl[off] + rl[off + 1]) + wr * (rr[off] + rr[off + 1]);
            contrib_k += wl * (kl[off] + kl[off + 1]) + wr * (kr[off] + kr[off + 1]);
        }
    }

    // Block reduction (deterministic; no float atomics).
    __shared__ float s0[BLOCK_THREADS];
    __shared__ float s1[BLOCK_THREADS];
    __shared__ float s2[BLOCK_THREADS];
    s0[tid] = contrib_r; s1[tid] = contrib_k; s2[tid] = contrib_c;
    __syncthreads();
    for (int s = BLOCK_THREADS / 2; s > 0; s >>= 1) {
        if (tid < s) {
            s0[tid] += s0[tid + s];
            s1[tid] += s1[tid + s];
            s2[tid] += s2[tid + s];
        }
        __syncthreads();
    }
    if (tid == 0) {
        ws[(size_t)blockIdx.x * 3 + 0] = s0[0];
        ws[(size_t)blockIdx.x * 3 + 1] = s1[0];
        ws[(size_t)blockIdx.x * 3 + 2] = s2[0];
    }
}

__global__ __launch_bounds__(256)
void pose_loss_stage2(const float* __restrict__ ws, float* __restrict__ out, int nparts) {
    const int tid = threadIdx.x;
    float r = 0.0f, k = 0.0f, c = 0.0f;
    for (int i = tid; i < nparts; i += 256) {
        r += ws[(size_t)i * 3 + 0];
        k += ws[(size_t)i * 3 + 1];
        c += ws[(size_t)i * 3 + 2];
    }
    __shared__ float s0[256];
    __shared__ float s1[256];
    __shared__ float s2[256];
    s0[tid] = r; s1[tid] = k; s2[tid] = c;
    __syncthreads();
    for (int s = 128; s > 0; s >>= 1) {
        if (tid < s) {
            s0[tid] += s0[tid + s];
            s1[tid] += s1[tid + s];
            s2[tid] += s2[tid + s];
        }
        __syncthreads();
    }
    if (tid == 0) {
        out[0] = s0[0] / 154.0f;   // recon_loss_all / 154
        out[1] = s1[0];            // KLD_loss_all
        out[2] = s2[0];            // count
    }
}

extern "C" void kernel_launch(void* const* d_in, const int* in_sizes, int n_in,
                              void* d_out, int out_size, void* d_ws, size_t ws_size,
                              hipStream_t stream) {
    const float* rl = (const float*)d_in[0];
    const float* rr = (const float*)d_in[1];
    const float* kl = (const float*)d_in[2];
    const float* kr = (const float*)d_in[3];
    const int*   lr = (const int*)d_in[4];
    const int*   fp = (const int*)d_in[5];
    float* out = (float*)d_out;
    float* ws  = (float*)d_ws;

    const int nrows  = in_sizes[4];                                  // B = 65536
    const int blocks = (nrows + ROWS_PER_BLOCK - 1) / ROWS_PER_BLOCK; // 1024

    pose_loss_stage1<<<blocks, BLOCK_THREADS, 0, stream>>>(rl, rr, kl, kr, lr, fp, ws, nrows);
    pose_loss_stage2<<<1, 256, 0, stream>>>(ws, out, blocks);
}